// GCN_v1_34772055229087
// MI455X (gfx1250) — compile-verified
//
#include <hip/hip_runtime.h>
#include <hip/hip_bf16.h>

typedef __attribute__((ext_vector_type(2))) float v2f;
typedef __attribute__((ext_vector_type(8))) float v8f;

#define HDIM 128
#define NWAVES 8

// ---------------- utility fills ----------------
__global__ void gcn_fill_kernel(float* __restrict__ p, float v, int n) {
    int i = blockIdx.x * blockDim.x + threadIdx.x;
    if (i < n) p[i] = v;
}

// ---------------- degree / normalization ----------------
__global__ void gcn_deg_accum_kernel(const int* __restrict__ dst, float* __restrict__ deg, int E) {
    int e = blockIdx.x * blockDim.x + threadIdx.x;
    if (e < E) atomicAdd(&deg[dst[e]], 1.0f);
}

__global__ void gcn_rsqrt_kernel(float* __restrict__ deg, int n) {
    int i = blockIdx.x * blockDim.x + threadIdx.x;
    if (i < n) deg[i] = rsqrtf(deg[i]);   // in-place: deg -> dis
}

// ---------------- WMMA fp32 GEMM: Y[M x NC] = X[M x 128] @ W[128 x NC] (+bias) ----------------
// grid.x = ceil(M/16); block = 256 (8 waves); wave w handles column tiles w, w+8, ...
// B loads are UNCONDITIONAL with a clamped column: out-of-range columns only feed
// accumulator lanes whose stores are masked, so no zeroing/EXEC-masking is needed.
__global__ __launch_bounds__(32 * NWAVES)
void gcn_gemm128_wmma_kernel(const float* __restrict__ X, const float* __restrict__ W,
                             const float* __restrict__ bias, float* __restrict__ Y,
                             int M, int NC) {
    __shared__ float sX[16 * HDIM];   // 8 KB row tile, shared by all 8 waves
    const int lane = threadIdx.x & 31;
    const int wave = threadIdx.x >> 5;
    const int row0 = blockIdx.x << 4;
    const bool fullTile = (row0 + 16 <= M);

    // stage X tile with 128-bit transfers (16 rows x 32 float4)
    {
        const float4* Xv = (const float4*)X;
        float4* sXv = (float4*)sX;
        for (int i = threadIdx.x; i < 16 * (HDIM / 4); i += 32 * NWAVES) {
            int r = i >> 5, c4 = i & 31;
            int gr = row0 + r;
            float4 v = make_float4(0.f, 0.f, 0.f, 0.f);
            if (gr < M) v = Xv[(size_t)gr * (HDIM / 4) + c4];
            sXv[i] = v;
        }
    }
    __syncthreads();

    const int nTiles = (NC + 15) >> 4;
    const int m  = lane & 15;     // A-fragment row
    const int kh = lane >> 4;     // K half: lanes 0-15 -> K=0,1 ; lanes 16-31 -> K=2,3
    const float* sA = sX + m * HDIM + 2 * kh;

    for (int t = wave; t < nTiles; t += NWAVES) {
        const int col0 = t << 4;
        const int gc   = col0 + (lane & 15);
        const int cc   = (gc < NC) ? gc : (NC - 1);          // clamp, no masking
        const float* wp = W + (size_t)(2 * kh) * NC + cc;    // row (2*kh), walk K by stride NC
        v8f acc = {};
#pragma unroll 8
        for (int k = 0; k < HDIM; k += 4) {
            v2f a, b;
            a.x = sA[k + 0];
            a.y = sA[k + 1];
            b.x = wp[(size_t)(k + 0) * NC];
            b.y = wp[(size_t)(k + 1) * NC];
            acc = __builtin_amdgcn_wmma_f32_16x16x4_f32(
                false, a, false, b, (short)0, acc, false, false);
        }
        // D layout: VGPR r -> row r (lanes 0-15) / row r+8 (lanes 16-31), col = lane&15
        const int dmb = (lane >> 4) << 3;
        if (gc < NC) {
            float bv = bias ? bias[gc] : 0.0f;
            float* yp = Y + (size_t)(row0 + dmb) * NC + gc;
            if (fullTile) {
                // uniform fast path: straight-line, clause-able stores
#pragma unroll
                for (int r = 0; r < 8; ++r) yp[(size_t)r * NC] = acc[r] + bv;
            } else {
#pragma unroll
                for (int r = 0; r < 8; ++r) {
                    if (row0 + dmb + r < M) yp[(size_t)r * NC] = acc[r] + bv;
                }
            }
        }
    }
}

// ---------------- edge scatter: one wave32 per edge, 512B row via float4 lanes ----------------
__global__ __launch_bounds__(256)
void gcn_scatter_kernel(const float* __restrict__ xw, const int* __restrict__ src,
                        const int* __restrict__ dst, const float* __restrict__ dis,
                        float* __restrict__ agg, int E) {
    int wid  = (blockIdx.x * blockDim.x + threadIdx.x) >> 5;
    int lane = threadIdx.x & 31;
    if (wid >= E) return;
    int s = src[wid], d = dst[wid];
    float nrm = dis[s] * dis[d];
    float4 v = ((const float4*)(xw + (size_t)s * HDIM))[lane];
    float* out = agg + (size_t)d * HDIM + lane * 4;
    atomicAdd(out + 0, v.x * nrm);
    atomicAdd(out + 1, v.y * nrm);
    atomicAdd(out + 2, v.z * nrm);
    atomicAdd(out + 3, v.w * nrm);
}

// ---------------- fused self-loop + bias + ReLU + LayerNorm (one wave per node) ----------------
__global__ __launch_bounds__(256)
void gcn_finalize_ln_kernel(const float* __restrict__ agg, const float* __restrict__ xw,
                            const float* __restrict__ dis, const float* __restrict__ b,
                            const float* __restrict__ g, const float* __restrict__ be,
                            float* __restrict__ out, int N) {
    int node = (blockIdx.x * blockDim.x + threadIdx.x) >> 5;
    int lane = threadIdx.x & 31;
    if (node >= N) return;
    float d2 = dis[node]; d2 *= d2;                 // self-loop norm = dis[i]^2
    const size_t base = (size_t)node * HDIM;
    float v[4], sum = 0.0f;
#pragma unroll
    for (int i = 0; i < 4; ++i) {
        int f = lane + 32 * i;
        float t = agg[base + f] + xw[base + f] * d2 + b[f];
        t = fmaxf(t, 0.0f);                         // ReLU before LN
        v[i] = t; sum += t;
    }
#pragma unroll
    for (int o = 16; o > 0; o >>= 1) sum += __shfl_xor(sum, o, 32);
    float mu = sum * (1.0f / HDIM);
    float var = 0.0f;
#pragma unroll
    for (int i = 0; i < 4; ++i) { float dd = v[i] - mu; var += dd * dd; }
#pragma unroll
    for (int o = 16; o > 0; o >>= 1) var += __shfl_xor(var, o, 32);
    float rs = rsqrtf(var * (1.0f / HDIM) + 1e-5f);
#pragma unroll
    for (int i = 0; i < 4; ++i) {
        int f = lane + 32 * i;
        out[base + f] = (v[i] - mu) * rs * g[f] + be[f];
    }
}

// ---------------- layer-2 finalize: emb (pre-ReLU) to d_out + ReLU copy for MLP ----------------
__global__ void gcn_finalize_emb_kernel(const float* __restrict__ agg, const float* __restrict__ xw,
                                        const float* __restrict__ dis, const float* __restrict__ b,
                                        float* __restrict__ emb, float* __restrict__ relu_out, int N) {
    int i = blockIdx.x * blockDim.x + threadIdx.x;
    if (i >= N * HDIM) return;
    int node = i >> 7, f = i & (HDIM - 1);
    float d = dis[node];
    float t = agg[i] + xw[i] * d * d + b[f];
    emb[i] = t;
    relu_out[i] = fmaxf(t, 0.0f);
}

// ---------------- log-softmax over C=40 classes (one wave per node) ----------------
__global__ __launch_bounds__(256)
void gcn_logsoftmax_kernel(const float* __restrict__ logits, float* __restrict__ out, int N, int C) {
    int node = (blockIdx.x * blockDim.x + threadIdx.x) >> 5;
    int lane = threadIdx.x & 31;
    if (node >= N) return;
    const size_t base = (size_t)node * C;
    float x0 = (lane < C)      ? logits[base + lane]      : -1e30f;
    float x1 = (lane + 32 < C) ? logits[base + lane + 32] : -1e30f;
    float mx = fmaxf(x0, x1);
#pragma unroll
    for (int o = 16; o > 0; o >>= 1) mx = fmaxf(mx, __shfl_xor(mx, o, 32));
    float s = 0.0f;
    if (lane < C)      s += expf(x0 - mx);
    if (lane + 32 < C) s += expf(x1 - mx);
#pragma unroll
    for (int o = 16; o > 0; o >>= 1) s += __shfl_xor(s, o, 32);
    float lse = mx + logf(s);
    if (lane < C)      out[base + lane]      = x0 - lse;
    if (lane + 32 < C) out[base + lane + 32] = x1 - lse;
}

// ---------------- host orchestration ----------------
extern "C" void kernel_launch(void* const* d_in, const int* in_sizes, int n_in,
                              void* d_out, int out_size, void* d_ws, size_t ws_size,
                              hipStream_t stream) {
    const float* x     = (const float*)d_in[0];
    const int*   ei    = (const int*)d_in[1];
    const float* W1    = (const float*)d_in[2];
    const float* b1    = (const float*)d_in[3];
    const float* W2    = (const float*)d_in[4];
    const float* b2    = (const float*)d_in[5];
    const float* W3    = (const float*)d_in[6];
    const float* b3    = (const float*)d_in[7];
    const float* ln1g  = (const float*)d_in[8];
    const float* ln1b  = (const float*)d_in[9];
    const float* ln2g  = (const float*)d_in[10];
    const float* ln2b  = (const float*)d_in[11];
    const float* mpW1  = (const float*)d_in[12];
    const float* mpb1  = (const float*)d_in[13];
    const float* mpW2  = (const float*)d_in[14];
    const float* mpb2  = (const float*)d_in[15];

    const int N = in_sizes[0] / HDIM;
    const int E = in_sizes[1] / 2;
    const int C = in_sizes[15];
    const int* src = ei;
    const int* dst = ei + E;

    float* dis  = (float*)d_ws;             // N
    float* bufA = dis  + N;                 // N*128  (XW)
    float* bufB = bufA + (size_t)N * HDIM;  // N*128  (agg / logits)
    float* bufC = bufB + (size_t)N * HDIM;  // N*128  (h / relu(emb))
    float* emb  = (float*)d_out;            // N*128
    float* lsm  = emb + (size_t)N * HDIM;   // N*C

    const int NH = N * HDIM;
    const int rowTiles = (N + 15) / 16;
    dim3 blk256(256);
    dim3 gNH((NH + 255) / 256), gN((N + 255) / 256), gE((E + 255) / 256);
    dim3 gWaveN((N * 32 + 255) / 256);          // one wave per node
    dim3 gWaveE(((size_t)E * 32 + 255) / 256);  // one wave per edge

    // degree -> dis
    gcn_fill_kernel<<<gN, blk256, 0, stream>>>(dis, 1.0f, N);           // self-loop
    gcn_deg_accum_kernel<<<gE, blk256, 0, stream>>>(dst, dis, E);
    gcn_rsqrt_kernel<<<gN, blk256, 0, stream>>>(dis, N);

    const float* h = x;
    const float* Ws[3]  = {W1, W2, W3};
    const float* bs[3]  = {b1, b2, b3};
    const float* lg[2]  = {ln1g, ln2g};
    const float* lb[2]  = {ln1b, ln2b};

    for (int layer = 0; layer < 3; ++layer) {
        gcn_gemm128_wmma_kernel<<<rowTiles, blk256, 0, stream>>>(h, Ws[layer], nullptr, bufA, N, HDIM);
        gcn_fill_kernel<<<gNH, blk256, 0, stream>>>(bufB, 0.0f, NH);
        gcn_scatter_kernel<<<gWaveE, blk256, 0, stream>>>(bufA, src, dst, dis, bufB, E);
        if (layer < 2) {
            gcn_finalize_ln_kernel<<<gWaveN, blk256, 0, stream>>>(
                bufB, bufA, dis, bs[layer], lg[layer], lb[layer], bufC, N);
            h = bufC;
        } else {
            gcn_finalize_emb_kernel<<<gNH, blk256, 0, stream>>>(
                bufB, bufA, dis, bs[layer], emb, bufC, N);
        }
    }

    // post-MP MLP head: hid = relu(emb)@mpW1 + mpb1 ; logits = hid@mpW2 + mpb2
    gcn_gemm128_wmma_kernel<<<rowTiles, blk256, 0, stream>>>(bufC, mpW1, mpb1, bufA, N, HDIM);
    gcn_gemm128_wmma_kernel<<<rowTiles, blk256, 0, stream>>>(bufA, mpW2, mpb2, bufB, N, C);
    gcn_logsoftmax_kernel<<<gWaveN, blk256, 0, stream>>>(bufB, lsm, N, C);
}